// mgn_ODERNN_60224031424822
// MI455X (gfx1250) — compile-verified
//
#include <hip/hip_runtime.h>
#include <hip/hip_bf16.h>

// Problem constants (match reference)
#define BATCH 8192
#define NCH   128
#define DDIM  16
#define NOBS  4096
#define HDIM  (NCH * DDIM)   // 2048
#define LSTR  18             // LDS row stride (floats): keeps b64 alignment, conflict-free

typedef __attribute__((ext_vector_type(2))) float v2f;
typedef __attribute__((ext_vector_type(8))) float v8f;

// ---------------------------------------------------------------------------
// V_WMMA_F32_16X16X4_F32:  D(16x16,f32) = A(16x4,f32) * B(4x16,f32) + C
// A: 2 VGPRs (lanes 0-15: M=lane, K=0/1; lanes 16-31: M=lane-16, K=2/3)
// B: 2 VGPRs (VGPR j: lanes 0-15 K=j, lanes 16-31 K=j+2; N=lane&15)
// C/D: 8 VGPRs (VGPR r: lanes 0-15 M=r, lanes 16-31 M=r+8; N=lane&15)
// ---------------------------------------------------------------------------
__device__ __forceinline__ v8f wmma4(v2f a, v2f b, v8f c) {
  return __builtin_amdgcn_wmma_f32_16x16x4_f32(
      /*neg_a=*/false, a, /*neg_b=*/false, b,
      /*c_mod=*/(short)0, c, /*reuse_a=*/false, /*reuse_b=*/false);
}

// Full 16x16 @ 16x16 as 4 chained K=4 WMMAs
__device__ __forceinline__ v8f mm16(const v2f a[4], const v2f b[4]) {
  v8f acc = {};
#pragma unroll
  for (int kk = 0; kk < 4; ++kk) acc = wmma4(a[kk], b[kk], acc);
  return acc;
}

// Load a 16x16 row-major weight matrix into B-operand layout.
__device__ __forceinline__ void loadB(const float* __restrict__ U, int col, int hi,
                                      v2f b[4]) {
#pragma unroll
  for (int kk = 0; kk < 4; ++kk) {
    const int k0 = 4 * kk + 2 * hi;
    b[kk][0] = U[k0 * 16 + col];
    b[kk][1] = U[(k0 + 1) * 16 + col];
  }
}

// Wave-local LDS phase separator: LDS ops from one wave complete in order
// (DScnt), so draining DScnt + blocking compiler reordering is sufficient.
__device__ __forceinline__ void wsync() {
  __builtin_amdgcn_wave_barrier();
  asm volatile("s_wait_dscnt 0" ::: "memory");
  __builtin_amdgcn_wave_barrier();
}

// A-operand register tile -> wave-private LDS tile (b64 stores, 8B aligned)
__device__ __forceinline__ void a_to_lds(float* __restrict__ sm, int lane,
                                         const v2f a[4]) {
  const int row = lane & 15;
  const int koff = (lane >> 4) * 2;
#pragma unroll
  for (int kk = 0; kk < 4; ++kk)
    *(v2f*)(sm + row * LSTR + 4 * kk + koff) = a[kk];
}

// C/D-layout register tile -> wave-private LDS tile
__device__ __forceinline__ void cd_to_lds(float* __restrict__ sm, int hi, int col,
                                          const float v[8]) {
#pragma unroll
  for (int r = 0; r < 8; ++r) sm[(r + 8 * hi) * LSTR + col] = v[r];
}

// LDS tile -> C/D-layout registers
__device__ __forceinline__ void lds_to_cd(const float* __restrict__ sm, int hi,
                                          int col, float v[8]) {
#pragma unroll
  for (int r = 0; r < 8; ++r) v[r] = sm[(r + 8 * hi) * LSTR + col];
}

// LDS tile -> A-operand layout registers (b64 loads)
__device__ __forceinline__ void lds_to_a(const float* __restrict__ sm, int lane,
                                         v2f a[4]) {
  const int row = lane & 15;
  const int koff = (lane >> 4) * 2;
#pragma unroll
  for (int kk = 0; kk < 4; ++kk)
    a[kk] = *(const v2f*)(sm + row * LSTR + 4 * kk + koff);
}

__device__ __forceinline__ float tanh_fast(float x) {
  x = fminf(fmaxf(x, -15.0f), 15.0f);
  const float e = __expf(2.0f * x);
  return (e - 1.0f) / (e + 1.0f);
}
__device__ __forceinline__ float sigmoid_fast(float x) {
  x = fminf(fmaxf(x, -30.0f), 30.0f);
  return 1.0f / (1.0f + __expf(-x));
}

// ---------------------------------------------------------------------------
// Kernel 1: ODE step for all BATCH rows.
//   out = mgn_h + dt * tanh(tanh(h@U1 + b1)@U2 + b2)
// One wave per (16 batch rows x 1 channel) tile; 4 waves per block.
// The h tile is read from HBM exactly once (A layout); the C/D view is
// re-materialized through the wave-private LDS tile.
// ---------------------------------------------------------------------------
__global__ __launch_bounds__(128) void ode_kernel(
    const float* __restrict__ mgn_h, const float* __restrict__ delta_t,
    const float* __restrict__ U1, const float* __restrict__ U2,
    const float* __restrict__ b1, const float* __restrict__ b2,
    float* __restrict__ out) {
  __shared__ float lds[4][16 * LSTR];
  const int lane = threadIdx.x & 31;
  const int wid = threadIdx.x >> 5;
  const int tile = blockIdx.x * 4 + wid;     // 65536 tiles
  const int c = tile >> 9;                   // tile / (BATCH/16)
  const int b0 = (tile & 511) * 16;
  const int hi = lane >> 4;
  const int col = lane & 15;
  const int row = lane & 15;
  const int koff = hi * 2;

  const float dt = delta_t[0];
  const float* hbase = mgn_h + (size_t)b0 * HDIM + c * DDIM;
  float* sm = lds[wid];

  // H in A-operand layout: one coalesced pass over the tile, no redundancy.
  v2f hA[4];
#pragma unroll
  for (int kk = 0; kk < 4; ++kk) {
    const float2 t = *(const float2*)(hbase + (size_t)row * HDIM + 4 * kk + koff);
    hA[kk][0] = t.x;
    hA[kk][1] = t.y;
  }
  // Stage tile in LDS and pull the C/D view from there (no 2nd HBM read).
  a_to_lds(sm, lane, hA);

  v2f B1[4], B2[4];
  loadB(U1 + c * 256, col, hi, B1);
  loadB(U2 + c * 256, col, hi, B2);
  const float b1v = b1[c * DDIM + col];
  const float b2v = b2[c * DDIM + col];

  wsync();
  float hCD[8];
  lds_to_cd(sm, hi, col, hCD);

  // layer 1 (A operand still live in registers)
  v8f a1 = mm16(hA, B1);
  float t1[8];
#pragma unroll
  for (int r = 0; r < 8; ++r) t1[r] = tanh_fast(a1[r] + b1v);

  // re-layout t1 (C/D) -> A via LDS (in-order DS makes read-then-overwrite safe)
  wsync();
  cd_to_lds(sm, hi, col, t1);
  wsync();
  v2f dA[4];
  lds_to_a(sm, lane, dA);

  // layer 2 + residual
  v8f a2 = mm16(dA, B2);
#pragma unroll
  for (int r = 0; r < 8; ++r) {
    const float dh = tanh_fast(a2[r] + b2v);
    const float h = hCD[r] + dt * dh;
    out[(size_t)(b0 + r + 8 * hi) * HDIM + c * DDIM + col] = h;
  }
}

// ---------------------------------------------------------------------------
// Kernel 2: GRU update for NOBS observed rows, gather/scatter via i_obs.
//   r = sig(x*Wr + hs@Ur + br); z = sig(x*Wz + hs@Uz + bz)
//   h~ = tanh(x*Wh + (r*hs)@Uh + bh); hn = z*hs + (1-z)*h~
//   out[i_obs] = hs*(1-m) + hn*m
// hs gathered from HBM once (A layout); C/D view comes from LDS.
// ---------------------------------------------------------------------------
__global__ __launch_bounds__(128) void gru_kernel(
    const float* __restrict__ X, const float* __restrict__ Mo,
    const int* __restrict__ i_obs,
    const float* __restrict__ Wr, const float* __restrict__ Wz,
    const float* __restrict__ Wh,
    const float* __restrict__ Ur, const float* __restrict__ Uz,
    const float* __restrict__ Uh,
    const float* __restrict__ br, const float* __restrict__ bz,
    const float* __restrict__ bh,
    float* __restrict__ out) {
  __shared__ float lds[4][16 * LSTR];
  const int lane = threadIdx.x & 31;
  const int wid = threadIdx.x >> 5;
  const int tile = blockIdx.x * 4 + wid;     // 32768 tiles
  const int c = tile >> 8;                   // tile / (NOBS/16)
  const int b0 = (tile & 255) * 16;
  const int hi = lane >> 4;
  const int col = lane & 15;
  const int row = lane & 15;
  const int koff = hi * 2;
  float* sm = lds[wid];

  // gathered h-row indices (scatter targets)
  int rowsCD[8];
#pragma unroll
  for (int r = 0; r < 8; ++r) rowsCD[r] = i_obs[b0 + r + 8 * hi];
  const int rowA = i_obs[b0 + row];

  // hs in A layout (single gathered HBM pass over the tile)
  v2f hA[4];
#pragma unroll
  for (int kk = 0; kk < 4; ++kk) {
    const float2 t =
        *(const float2*)(out + (size_t)rowA * HDIM + c * DDIM + 4 * kk + koff);
    hA[kk][0] = t.x;
    hA[kk][1] = t.y;
  }
  a_to_lds(sm, lane, hA);

  v2f BR[4], BZ[4], BH[4];
  loadB(Ur + c * 256, col, hi, BR);
  loadB(Uz + c * 256, col, hi, BZ);
  loadB(Uh + c * 256, col, hi, BH);
  const float brv = br[c * DDIM + col];
  const float bzv = bz[c * DDIM + col];
  const float bhv = bh[c * DDIM + col];
  const float wrv = Wr[c * DDIM + col];
  const float wzv = Wz[c * DDIM + col];
  const float whv = Wh[c * DDIM + col];

  // per observed-row scalars (obs index, not gathered index)
  float xv[8], mv[8];
#pragma unroll
  for (int r = 0; r < 8; ++r) {
    const int rr = b0 + r + 8 * hi;
    xv[r] = X[(size_t)rr * NCH + c];
    mv[r] = Mo[(size_t)rr * NCH + c];
  }

  wsync();
  float h[8];
  lds_to_cd(sm, hi, col, h);

  // gates r, z (A operand live in registers)
  const v8f aR = mm16(hA, BR);
  const v8f aZ = mm16(hA, BZ);
  float rg[8], zg[8];
#pragma unroll
  for (int r = 0; r < 8; ++r) {
    rg[r] = sigmoid_fast(xv[r] * wrv + aR[r] + brv);
    zg[r] = sigmoid_fast(xv[r] * wzv + aZ[r] + bzv);
  }

  // (r * hs) -> A layout via LDS
  float rh[8];
#pragma unroll
  for (int r = 0; r < 8; ++r) rh[r] = rg[r] * h[r];
  wsync();
  cd_to_lds(sm, hi, col, rh);
  wsync();
  v2f rhA[4];
  lds_to_a(sm, lane, rhA);

  // candidate + blend + mask, scatter
  const v8f aH = mm16(rhA, BH);
#pragma unroll
  for (int r = 0; r < 8; ++r) {
    const float ht = tanh_fast(xv[r] * whv + aH[r] + bhv);
    const float hn = zg[r] * h[r] + (1.0f - zg[r]) * ht;
    const float o = h[r] * (1.0f - mv[r]) + hn * mv[r];
    out[(size_t)rowsCD[r] * HDIM + c * DDIM + col] = o;
  }
}

extern "C" void kernel_launch(void* const* d_in, const int* in_sizes, int n_in,
                              void* d_out, int out_size, void* d_ws,
                              size_t ws_size, hipStream_t stream) {
  const float* mgn_h = (const float*)d_in[0];
  const float* X     = (const float*)d_in[1];
  const float* Mo    = (const float*)d_in[2];
  const float* dt    = (const float*)d_in[3];
  const int*   iobs  = (const int*)d_in[4];
  const float* Wr    = (const float*)d_in[5];
  const float* Wz    = (const float*)d_in[6];
  const float* Wh    = (const float*)d_in[7];
  const float* Ur    = (const float*)d_in[8];
  const float* Uz    = (const float*)d_in[9];
  const float* Uh    = (const float*)d_in[10];
  const float* br    = (const float*)d_in[11];
  const float* bz    = (const float*)d_in[12];
  const float* bh    = (const float*)d_in[13];
  const float* U1    = (const float*)d_in[14];
  const float* U2    = (const float*)d_in[15];
  const float* b1    = (const float*)d_in[16];
  const float* b2    = (const float*)d_in[17];
  float* out = (float*)d_out;

  // Kernel 1: (BATCH/16)*NCH tiles / 4 waves per block = 16384 blocks
  ode_kernel<<<(BATCH / 16) * NCH / 4, 128, 0, stream>>>(mgn_h, dt, U1, U2, b1,
                                                         b2, out);
  // Kernel 2: (NOBS/16)*NCH tiles / 4 = 8192 blocks
  gru_kernel<<<(NOBS / 16) * NCH / 4, 128, 0, stream>>>(
      X, Mo, iobs, Wr, Wz, Wh, Ur, Uz, Uh, br, bz, bh, out);
}